// PyrmaidCostVolume_31147102830791
// MI455X (gfx1250) — compile-verified
//
#include <hip/hip_runtime.h>
#include <stdint.h>

// Shapes fixed by setup_inputs(): B=4, D=256, H=128, W=256, 3 levels, 9 samples.
#define Bz    4
#define D0    256
#define Hh    128
#define Ww    256
#define HW    (Hh * Ww)          // 32768
#define NPIX  (Bz * HW)          // 131072
#define NSAMP 9
#define NLVL  3
#define OUTC  (NLVL * NSAMP)     // 27

// ---------------------------------------------------------------------------
// Avg-pool k=2 s=2 along D: dst[b,d,h,w] = 0.5*(src[b,2d,h,w]+src[b,2d+1,h,w])
// Fully coalesced streaming (contiguous in w across lanes). Destination stores
// are regular-temporal on purpose: p1/p2 must stay L2-resident for the sampler.
// ---------------------------------------------------------------------------
__global__ void pool2x_kernel(const float* __restrict__ src,
                              float* __restrict__ dst,
                              int Dd /* dst D */) {
    int idx = blockIdx.x * blockDim.x + threadIdx.x;
    int n = Bz * Dd * HW;
    if (idx >= n) return;
    int hw = idx % HW;
    int t  = idx / HW;
    int d  = t % Dd;
    int b  = t / Dd;
    const float* s = src + ((size_t)(b * (2 * Dd) + 2 * d) * HW + hw);
    dst[idx] = (s[0] + s[HW]) * 0.5f;
}

// ---------------------------------------------------------------------------
// Sampler: one thread per pixel, produces all 27 output channels.
// POOLED=true reads materialized pooled volumes p1/p2; POOLED=false pools
// on the fly from the raw cost volume (workspace-too-small fallback).
// Output stores are non-temporal: written once, never re-read -> keep the
// 192MB L2 for the gather-hot cost volumes instead.
// ---------------------------------------------------------------------------
template <bool POOLED>
__global__ __launch_bounds__(256)
void sampler_kernel(const float* __restrict__ cv,
                    const float* __restrict__ disp,
                    const int* __restrict__ radius,
                    const float* __restrict__ p1,
                    const float* __restrict__ p2,
                    float* __restrict__ out) {
    __shared__ float sdisp[256];
    const int tid = threadIdx.x;
    const int pix = blockIdx.x * 256 + tid;   // grid exactly covers NPIX

    // --- CDNA5 async path: stage this block's 256 contiguous disparities
    // into LDS via the async-to-LDS unit (tracked with ASYNCcnt). Each lane
    // copies and later reads only its own slot, so s_wait_asynccnt suffices
    // (no cross-wave barrier needed).
    {
        unsigned lds_addr = (unsigned)(uintptr_t)(&sdisp[tid]); // low 32b of flat = LDS offset
        unsigned voff = (unsigned)(tid * 4);
        unsigned long long gbase =
            (unsigned long long)(uintptr_t)(disp + (size_t)blockIdx.x * 256u);
        asm volatile("global_load_async_to_lds_b32 %0, %1, %2"
                     :
                     : "v"(lds_addr), "v"(voff), "s"(gbase)
                     : "memory");
        asm volatile("s_wait_asynccnt 0" ::: "memory");
    }
    const float dv = sdisp[tid];

    const float r = (float)radius[0];              // uniform scalar load
    const float interval = (2.0f * r) * (1.0f / (float)(NSAMP - 1)); // 2r/8

    const int b  = pix / HW;
    const int hw = pix % HW;
    float* outp = out + (size_t)b * OUTC * HW + hw;

#pragma unroll
    for (int lvl = 0; lvl < NLVL; ++lvl) {
        const int   Dl    = D0 >> lvl;
        const float ref   = dv / (float)(1 << lvl);
        const float lower = ref - r;
        const float dmax  = (float)(Dl - 1);

        const float* __restrict__ base =
            POOLED ? (lvl == 0 ? cv : (lvl == 1 ? p1 : p2)) : cv;
        const int Dbuf = POOLED ? Dl : D0;
        const float* __restrict__ bb = base + (size_t)b * Dbuf * HW + hw;

#pragma unroll
        for (int s = 0; s < NSAMP; ++s) {
            const float cand = lower + (float)s * interval;
            float cc = ceilf(cand);
            cc = fminf(fmaxf(cc, 0.0f), dmax);
            float fc = floorf(cand);
            fc = fminf(fmaxf(fc, 0.0f), dmax);
            const float fr = cc - cand;     // floor_rate (matches reference)
            const float cr = 1.0f - fr;     // ceil_rate
            const int ci = (int)cc;
            const int fi = (int)fc;

            float vc, vf;
            if (POOLED || lvl == 0) {
                vc = bb[(size_t)ci * HW];
                vf = bb[(size_t)fi * HW];
            } else if (lvl == 1) {
                const float* c0 = bb + (size_t)(2 * ci) * HW;
                const float* f0 = bb + (size_t)(2 * fi) * HW;
                vc = (c0[0] + c0[HW]) * 0.5f;
                vf = (f0[0] + f0[HW]) * 0.5f;
            } else { // lvl == 2: mean-of-means to match reference rounding
                const float* c0 = bb + (size_t)(4 * ci) * HW;
                const float* f0 = bb + (size_t)(4 * fi) * HW;
                vc = ((c0[0] + c0[HW]) * 0.5f + (c0[2 * HW] + c0[3 * HW]) * 0.5f) * 0.5f;
                vf = ((f0[0] + f0[HW]) * 0.5f + (f0[2 * HW] + f0[3 * HW]) * 0.5f) * 0.5f;
            }
            __builtin_nontemporal_store(vc * cr + vf * fr,
                                        outp + (size_t)(lvl * NSAMP + s) * HW);
        }
    }
}

// ---------------------------------------------------------------------------
extern "C" void kernel_launch(void* const* d_in, const int* in_sizes, int n_in,
                              void* d_out, int out_size, void* d_ws, size_t ws_size,
                              hipStream_t stream) {
    const float* cv   = (const float*)d_in[0];
    const float* disp = (const float*)d_in[1];
    const int*   rad  = (const int*)d_in[2];
    float* out = (float*)d_out;

    const size_t p1_elems = (size_t)Bz * (D0 / 2) * HW;  // 16.7M floats (64 MB)
    const size_t p2_elems = (size_t)Bz * (D0 / 4) * HW;  //  8.4M floats (32 MB)
    const size_t need = (p1_elems + p2_elems) * sizeof(float);

    if (ws_size >= need) {
        float* p1 = (float*)d_ws;
        float* p2 = p1 + p1_elems;
        const int n1 = (int)p1_elems;
        const int n2 = (int)p2_elems;
        pool2x_kernel<<<(n1 + 255) / 256, 256, 0, stream>>>(cv, p1, D0 / 2);
        pool2x_kernel<<<(n2 + 255) / 256, 256, 0, stream>>>(p1, p2, D0 / 4);
        sampler_kernel<true><<<NPIX / 256, 256, 0, stream>>>(cv, disp, rad, p1, p2, out);
    } else {
        sampler_kernel<false><<<NPIX / 256, 256, 0, stream>>>(cv, disp, rad,
                                                              nullptr, nullptr, out);
    }
}